// CCEncoding_48687749267804
// MI455X (gfx1250) — compile-verified
//
#include <hip/hip_runtime.h>

#define GRID_R 128
#define BLOCK 128          // 4 waves of 32
#define WPB (BLOCK / 32)

typedef float v4f __attribute__((ext_vector_type(4)));

__device__ __forceinline__ void fma4(v4f& a, float w, const v4f& f) {
  a.x = fmaf(w, f.x, a.x);
  a.y = fmaf(w, f.y, a.y);
  a.z = fmaf(w, f.z, a.z);
  a.w = fmaf(w, f.w, a.w);
}

__global__ __launch_bounds__(BLOCK) void trilerp_async_gather(
    const float* __restrict__ pts,
    const float* __restrict__ latents,
    float* __restrict__ out,
    int n)
{
  // [wave][corner 0..7][chunk 0..3][lane 0..31] : 16B each -> 16 KB per wave, 64 KB total
  __shared__ v4f stage[WPB * 8 * 4 * 32];

  const int tid  = blockIdx.x * BLOCK + threadIdx.x;
  const int lane = threadIdx.x & 31;
  const int wave = threadIdx.x >> 5;
  if (tid >= n) return;

  // ---- per-point coords & trilinear weights (pts streamed non-temporally) ----
  const float px = __builtin_nontemporal_load(pts + 3 * tid + 0);
  const float py = __builtin_nontemporal_load(pts + 3 * tid + 1);
  const float pz = __builtin_nontemporal_load(pts + 3 * tid + 2);

  const float sx = px * (float)(GRID_R - 2);
  const float sy = py * (float)(GRID_R - 2);
  const float sz = pz * (float)(GRID_R - 2);
  const float bx = floorf(sx), by = floorf(sy), bz = floorf(sz);
  const float fx = sx - bx, fy = sy - by, fz = sz - bz;
  const int ix = (int)bx, iy = (int)by, iz = (int)bz;

  int xs[2], ys[2], zs[2];
  xs[0] = min(max(ix, 0), GRID_R - 1);     xs[1] = min(max(ix + 1, 0), GRID_R - 1);
  ys[0] = min(max(iy, 0), GRID_R - 1);     ys[1] = min(max(iy + 1, 0), GRID_R - 1);
  zs[0] = min(max(iz, 0), GRID_R - 1);     zs[1] = min(max(iz + 1, 0), GRID_R - 1);
  float wx[2] = {1.f - fx, fx};
  float wy[2] = {1.f - fy, fy};
  float wz[2] = {1.f - fz, fz};

  uint32_t goff[8];   // byte offset of each corner's 64B feature row
  float    w[8];
#pragma unroll
  for (int a = 0; a < 2; ++a)
#pragma unroll
    for (int b = 0; b < 2; ++b)
#pragma unroll
      for (int c = 0; c < 2; ++c) {
        const int ci = 4 * a + 2 * b + c;   // matches _OFFSETS ordering (x,y,z bits)
        goff[ci] = (uint32_t)(xs[a] * (GRID_R * GRID_R) + ys[b] * GRID_R + zs[c]) * 64u;
        w[ci]    = wx[a] * wy[b] * wz[c];
      }

  // LDS byte offset of this lane's 16B slot (flat shared ptr low 32 bits = LDS offset)
  const uint32_t lds0 =
      (uint32_t)(size_t)(&stage[0]) + (uint32_t)(wave * 16384 + lane * 16);
  const unsigned long long lat = (unsigned long long)(const void*)latents;

  // ---- issue 32 async gathers: 8 corners x 4 b128 chunks; ASYNCcnt completes in order.
  //      Default RT temporal hint: keep the L2-resident latents table cached. ----
#pragma unroll
  for (int c = 0; c < 8; ++c) {
#pragma unroll
    for (int k = 0; k < 4; ++k) {
      const uint32_t loff = lds0 + (uint32_t)((c * 4 + k) * 512);
      const uint32_t voff = goff[c] + (uint32_t)(k * 16);
      asm volatile("global_load_async_to_lds_b128 %0, %1, %2"
                   :: "v"(loff), "v"(voff), "s"(lat)
                   : "memory");
    }
  }

  // ---- consume corners as their 4 chunks retire (staged s_wait_asynccnt) ----
  v4f a0 = {0.f, 0.f, 0.f, 0.f}, a1 = a0, a2 = a0, a3 = a0;
  const int base = wave * 1024 + lane;   // v4f index of [wave][corner0][chunk0][lane]

#define CORNER(C, WN)                                                     \
  do {                                                                    \
    asm volatile("s_wait_asynccnt " #WN ::: "memory");                    \
    const v4f f0 = stage[base + (C) * 128 + 0];                           \
    const v4f f1 = stage[base + (C) * 128 + 32];                          \
    const v4f f2 = stage[base + (C) * 128 + 64];                          \
    const v4f f3 = stage[base + (C) * 128 + 96];                          \
    const float wc = w[C];                                                \
    fma4(a0, wc, f0); fma4(a1, wc, f1); fma4(a2, wc, f2); fma4(a3, wc, f3);\
  } while (0)

  CORNER(0, 28); CORNER(1, 24); CORNER(2, 20); CORNER(3, 16);
  CORNER(4, 12); CORNER(5, 8);  CORNER(6, 4);  CORNER(7, 0);
#undef CORNER

  // ---- coalesced b128 non-temporal stores: don't evict latents from L2 ----
  v4f* o = (v4f*)(out + (size_t)tid * 16);
  __builtin_nontemporal_store(a0, o + 0);
  __builtin_nontemporal_store(a1, o + 1);
  __builtin_nontemporal_store(a2, o + 2);
  __builtin_nontemporal_store(a3, o + 3);
}

extern "C" void kernel_launch(void* const* d_in, const int* in_sizes, int n_in,
                              void* d_out, int out_size, void* d_ws, size_t ws_size,
                              hipStream_t stream) {
  const float* pts     = (const float*)d_in[0];
  const float* latents = (const float*)d_in[1];
  float* out = (float*)d_out;
  const int n = in_sizes[0] / 3;          // N_PTS
  const int grid = (n + BLOCK - 1) / BLOCK;
  trilerp_async_gather<<<grid, BLOCK, 0, stream>>>(pts, latents, out, n);
}